// FactorizedDenseAttention_16286515986792
// MI455X (gfx1250) — compile-verified
//
#include <hip/hip_runtime.h>

typedef __attribute__((ext_vector_type(2))) float v2f;
typedef __attribute__((ext_vector_type(8))) float v8f;

#define CHUNK 128

// qs[b][d] = sum_i q[b][i][d]  -- coalesced flat reduction, blockDim % D == 0
__global__ void k_qsum(const float* __restrict__ q, float* __restrict__ qs,
                       int L, int D) {
    __shared__ float red[256];
    int b = blockIdx.x;
    const float* qb = q + (size_t)b * L * D;
    int n = L * D;
    float acc = 0.f;
    for (int f = threadIdx.x; f < n; f += blockDim.x) acc += qb[f];
    red[threadIdx.x] = acc;
    __syncthreads();
    if ((int)threadIdx.x < D) {
        float ssum = 0.f;
        for (int t = threadIdx.x; t < (int)blockDim.x; t += D) ssum += red[t];
        qs[b * D + threadIdx.x] = ssum;
    }
}

// fa[b*L+i] = q[b,i,:] . W_a + b_a
__global__ void k_fa(const float* __restrict__ q, const float* __restrict__ W_a,
                     const float* __restrict__ b_a, float* __restrict__ fa,
                     int BL, int D) {
    int idx = blockIdx.x * blockDim.x + threadIdx.x;
    if (idx >= BL) return;
    const float* qr = q + (size_t)idx * D;
    float acc = b_a[0];
    #pragma unroll 8
    for (int d = 0; d < D; ++d) acc = fmaf(qr[d], W_a[d], acc);
    fa[idx] = acc;
}

// s[b*L+j] = qs[b,:] . W_b[:,j] + L*b_b[j]
__global__ void k_s(const float* __restrict__ qs, const float* __restrict__ W_b,
                    const float* __restrict__ b_b, float* __restrict__ s,
                    int L, int D, int BL) {
    int idx = blockIdx.x * blockDim.x + threadIdx.x;
    if (idx >= BL) return;
    int b = idx / L, j = idx % L;
    const float* qsb = qs + b * D;
    float acc = (float)L * b_b[j];
    for (int d = 0; d < D; ++d) acc = fmaf(qsb[d], W_b[(size_t)d * L + j], acc);
    s[idx] = acc;
}

// per-batch max/min of s
__global__ void k_minmax(const float* __restrict__ s, float* __restrict__ smax,
                         float* __restrict__ smin, int L) {
    __shared__ float shx[256], shn[256];
    int b = blockIdx.x;
    const float* sb = s + (size_t)b * L;
    float mx = -__builtin_inff(), mn = __builtin_inff();
    for (int j = threadIdx.x; j < L; j += blockDim.x) {
        float val = sb[j];
        mx = fmaxf(mx, val);
        mn = fminf(mn, val);
    }
    shx[threadIdx.x] = mx; shn[threadIdx.x] = mn;
    __syncthreads();
    for (int off = blockDim.x >> 1; off > 0; off >>= 1) {
        if ((int)threadIdx.x < off) {
            shx[threadIdx.x] = fmaxf(shx[threadIdx.x], shx[threadIdx.x + off]);
            shn[threadIdx.x] = fminf(shn[threadIdx.x], shn[threadIdx.x + off]);
        }
        __syncthreads();
    }
    if (threadIdx.x == 0) { smax[b] = shx[0]; smin[b] = shn[0]; }
}

// per-row softmax max m_i and reciprocal partition 1/Z_i; one wave32 per row
__global__ void k_rowmz(const float* __restrict__ fa, const float* __restrict__ s,
                        const float* __restrict__ smax, const float* __restrict__ smin,
                        float* __restrict__ m_out, float* __restrict__ zinv_out,
                        int L, int BL) {
    int row = (blockIdx.x * blockDim.x + threadIdx.x) >> 5;   // wave32
    int lane = threadIdx.x & 31;
    if (row >= BL) return;
    int b = row / L;
    float f = fa[row];
    float m = (f >= 0.f) ? f * smax[b] : f * smin[b];
    const float* sb = s + (size_t)b * L;
    float z = 0.f;
    for (int j = lane; j < L; j += 32) z += __expf(f * sb[j] - m);
    #pragma unroll
    for (int off = 16; off > 0; off >>= 1) z += __shfl_down(z, off, 32);
    if (lane == 0) { m_out[row] = m; zinv_out[row] = 1.f / z; }
}

// Fused: compute normalized attn tile (16 x CHUNK) in LDS, stream it to HBM
// with non-temporal stores, and accumulate attn @ v via V_WMMA_F32_16X16X4_F32
// using two independent accumulators to break the XDL D->C dependency chain.
// Block = 128 threads = 4 waves; wave w handles d-tile [16w, 16w+16).
__global__ void __launch_bounds__(128) k_attn_out(
    const float* __restrict__ v, const float* __restrict__ fa,
    const float* __restrict__ s, const float* __restrict__ m_arr,
    const float* __restrict__ zinv, float* __restrict__ out,
    float* __restrict__ attn, int L, int D) {
    __shared__ float sE[CHUNK][17];           // [j-in-chunk][row], padded stride 17
    __shared__ float sFA[16], sM[16], sZ[16];

    int tilesPerB = L >> 4;
    int b  = blockIdx.x / tilesPerB;
    int i0 = (blockIdx.x % tilesPerB) << 4;
    int tid  = threadIdx.x;
    int wv   = tid >> 5;
    int lane = tid & 31;
    int lcol = lane & 15;
    int khalf = (lane < 16) ? 0 : 2;          // ISA A/B striping: K {0,1} vs {2,3}
    int dbase = wv << 4;

    if (tid < 16) {
        int r = b * L + i0 + tid;
        sFA[tid] = fa[r]; sM[tid] = m_arr[r]; sZ[tid] = zinv[r];
    }
    __syncthreads();

    const float* sb = s + (size_t)b * L;
    const float* vb = v + (size_t)b * L * D;
    float* attn_base = attn + (size_t)b * L * L + (size_t)i0 * L + tid;

    v8f acc0 = {};
    v8f acc1 = {};
    for (int jc = 0; jc < L; jc += CHUNK) {
        float sj = sb[jc + tid];              // one column j = jc+tid per thread
        float* ap = attn_base + jc;
        #pragma unroll
        for (int r = 0; r < 16; ++r) {
            float e = __expf(sFA[r] * sj - sM[r]) * sZ[r];
            sE[tid][r] = e;
            __builtin_nontemporal_store(e, ap);    // coalesced over tid, NT far
            ap += L;
        }
        __syncthreads();
        // per-lane moving pointer into v: B[K][N=lcol] for K = j, j+1
        const float* vp = vb + (size_t)(jc + khalf) * D + dbase + lcol;
        #pragma unroll 4
        for (int k = 0; k < CHUNK; k += 8) {
            v2f a, bm;
            a.x = sE[k + khalf + 0][lcol];
            a.y = sE[k + khalf + 1][lcol];
            bm.x = vp[0];
            bm.y = vp[D];
            acc0 = __builtin_amdgcn_wmma_f32_16x16x4_f32(
                false, a, false, bm, (short)0, acc0, false, false);
            vp += 4 * D;
            a.x = sE[k + 4 + khalf + 0][lcol];
            a.y = sE[k + 4 + khalf + 1][lcol];
            bm.x = vp[0];
            bm.y = vp[D];
            acc1 = __builtin_amdgcn_wmma_f32_16x16x4_f32(
                false, a, false, bm, (short)0, acc1, false, false);
            vp += 4 * D;
        }
        __syncthreads();
    }

    int rofs = (lane < 16) ? 0 : 8;           // C/D layout: VGPR r -> rows r / r+8
    float* op = out + ((size_t)b * L + i0 + rofs) * D + dbase + lcol;
    #pragma unroll
    for (int r = 0; r < 8; ++r) {
        __builtin_nontemporal_store(acc0[r] + acc1[r], op);
        op += D;
    }
}

extern "C" void kernel_launch(void* const* d_in, const int* in_sizes, int n_in,
                              void* d_out, int out_size, void* d_ws, size_t ws_size,
                              hipStream_t stream) {
    const float* q   = (const float*)d_in[0];
    const float* v   = (const float*)d_in[1];
    const float* W_a = (const float*)d_in[2];
    const float* b_a = (const float*)d_in[3];
    const float* W_b = (const float*)d_in[4];
    const float* b_b = (const float*)d_in[5];

    int D  = in_sizes[2];            // W_a: [D,1]
    int L  = in_sizes[5];            // b_b: [L]
    int B  = in_sizes[0] / (L * D);  // q: [B,L,D]
    int BL = B * L;

    float* ws   = (float*)d_ws;
    float* fa   = ws;                 // B*L
    float* s    = fa + BL;            // B*L
    float* qs   = s + BL;             // B*D
    float* smax = qs + B * D;         // B
    float* smin = smax + B;           // B
    float* m    = smin + B;           // B*L
    float* zi   = m + BL;             // B*L

    float* out  = (float*)d_out;              // [B,L,D]
    float* attn = out + (size_t)B * L * D;    // [B,L,L]

    k_qsum<<<B, 256, 0, stream>>>(q, qs, L, D);
    k_fa<<<(BL + 255) / 256, 256, 0, stream>>>(q, W_a, b_a, fa, BL, D);
    k_s<<<(BL + 255) / 256, 256, 0, stream>>>(qs, W_b, b_b, s, L, D, BL);
    k_minmax<<<B, 256, 0, stream>>>(s, smax, smin, L);
    k_rowmz<<<(BL * 32 + 255) / 256, 256, 0, stream>>>(fa, s, smax, smin, m, zi, L, BL);
    k_attn_out<<<B * (L / 16), 128, 0, stream>>>(v, fa, s, m, zi, out, attn, L, D);
}